// simple_GCN_res_25366076850191
// MI455X (gfx1250) — compile-verified
//
#include <hip/hip_runtime.h>
#include <hip/hip_bf16.h>

// ---------------------------------------------------------------------------
// simple_GCN_res for MI455X (gfx1250, wave32)
//
// Bandwidth-bound problem: 5x [100k,64]x[64,64] GEMMs (4.1 GFLOP total) +
// 5x 1.1M-edge gather/scatter (~2.8 GB of mostly-L2 traffic). GEMMs use
// V_WMMA_F32_16X16X4_F32 (full fp32 precision; compute is not the
// bottleneck so no reason to drop precision). Scatter uses f32 global
// atomics; node buffers (3 x 25.6 MB) fit in the 192 MB L2.
// ---------------------------------------------------------------------------

typedef __attribute__((ext_vector_type(2))) float v2f;
typedef __attribute__((ext_vector_type(8))) float v8f;

#define HDIM 64
#define EPSV 1e-5f

// ---------------- degree / normalization ----------------------------------
__global__ void deg_init_kernel(float* __restrict__ deg, int n) {
    int i = blockIdx.x * blockDim.x + threadIdx.x;
    if (i < n) deg[i] = 1.0f;  // self-loop
}

__global__ void deg_count_kernel(const int* __restrict__ ei, float* __restrict__ deg, int E) {
    int i = blockIdx.x * blockDim.x + threadIdx.x;
    if (i < E) atomicAdd(&deg[ei[E + i]], 1.0f);  // dst row
}

__global__ void deg_rsqrt_kernel(float* __restrict__ deg, int n) {
    int i = blockIdx.x * blockDim.x + threadIdx.x;
    if (i < n) deg[i] = rsqrtf(deg[i]);  // in-place -> dinv
}

// ---------------- WMMA f32 GEMM: out[N,64] = A[N,64] @ W[64,64] ------------
// Block = 256 threads = 8 waves. Each wave computes a 16x64 output tile.
// A-block (128x64) and W (64x64) staged in LDS (48 KB).
__global__ void __launch_bounds__(256) gemm_kernel(const float* __restrict__ A,
                                                   const float* __restrict__ W,
                                                   float* __restrict__ out, int N) {
    __shared__ float sW[HDIM * HDIM];     // 16 KB
    __shared__ float sA[128 * HDIM];      // 32 KB
    const int tid = threadIdx.x;
    const int rowBase = blockIdx.x * 128;

    // stage W: 4096 floats = 1024 float4
    {
        const float4* Wv = (const float4*)W;
        float4* sWv = (float4*)sW;
        for (int i = tid; i < 1024; i += 256) sWv[i] = Wv[i];
    }
    // stage A block: 128 rows x 64 = 2048 float4 (zero-pad past N)
    {
        const float4* Av = (const float4*)A;
        float4* sAv = (float4*)sA;
        for (int i = tid; i < 2048; i += 256) {
            int row = rowBase + (i >> 4);  // 16 float4 per row
            float4 v = {0.f, 0.f, 0.f, 0.f};
            if (row < N) v = Av[(size_t)row * 16 + (i & 15)];
            sAv[i] = v;
        }
    }
    __syncthreads();

    const int wave  = tid >> 5;
    const int lane  = tid & 31;
    const int lrow  = lane & 15;   // M (for A) / N (for B) within tile
    const int khalf = lane >> 4;   // which K pair this half-wave holds
    const float* aRow = &sA[(wave * 16 + lrow) * HDIM];
    const int outRowBase = rowBase + wave * 16;

    for (int nt = 0; nt < 4; ++nt) {            // 4 x 16-column tiles
        v8f c = {};
        for (int kc = 0; kc < 16; ++kc) {       // K = 64 in chunks of 4
            const int k0 = kc * 4 + khalf * 2;
            v2f a, b;
            // A 16x4 f32 layout: lanes 0-15 K={0,1}, lanes 16-31 K={2,3}
            a.x = aRow[k0];
            a.y = aRow[k0 + 1];
            // B 4x16 f32 layout (mirrored): per lane col = lrow
            b.x = sW[(k0    ) * HDIM + nt * 16 + lrow];
            b.y = sW[(k0 + 1) * HDIM + nt * 16 + lrow];
            c = __builtin_amdgcn_wmma_f32_16x16x4_f32(
                    false, a, false, b, (short)0, c, false, false);
        }
        // C/D: VGPR v holds M = v + khalf*8, N = lrow
        for (int v = 0; v < 8; ++v) {
            int row = outRowBase + v + khalf * 8;
            if (row < N) out[(size_t)row * HDIM + nt * 16 + lrow] = c[v];
        }
    }
}

// ---------------- acc = t * dinv^2 + bias (self-loop + bias) ---------------
__global__ void init_acc_kernel(const float* __restrict__ t, const float* __restrict__ dinv,
                                const float* __restrict__ bias, float* __restrict__ acc,
                                int total) {
    int i = blockIdx.x * blockDim.x + threadIdx.x;
    if (i >= total) return;
    int node = i >> 6;
    int ch = i & 63;
    float d = dinv[node];
    acc[i] = t[i] * d * d + bias[ch];
}

// ---------------- edge scatter: acc[dst] += t[src] * dinv[src]*dinv[dst] ---
// One wave32 per edge; lane handles 2 channels (float2 gather + 2 atomics).
__global__ void __launch_bounds__(256) scatter_kernel(const float* __restrict__ t,
                                                      const int* __restrict__ ei,
                                                      const float* __restrict__ dinv,
                                                      float* __restrict__ acc, int E) {
    int e = blockIdx.x * 8 + (threadIdx.x >> 5);
    int lane = threadIdx.x & 31;
    if (e >= E) return;
    int s = ei[e];
    int d = ei[E + e];
    float w = dinv[s] * dinv[d];
    const float2* tv = (const float2*)(t + (size_t)s * HDIM);
    float2 v = tv[lane];
    float* ap = acc + (size_t)d * HDIM + lane * 2;
    atomicAdd(ap,     v.x * w);
    atomicAdd(ap + 1, v.y * w);
}

// ---------------- xnew = BN(ReLU(acc)) [+ res] -----------------------------
__global__ void post_kernel(const float* __restrict__ acc, const float* __restrict__ res,
                            const float* __restrict__ gam, const float* __restrict__ bet,
                            const float* __restrict__ mean, const float* __restrict__ var,
                            float* __restrict__ out, int total) {
    int i = blockIdx.x * blockDim.x + threadIdx.x;
    if (i >= total) return;
    int ch = i & 63;
    float v = fmaxf(acc[i], 0.0f);
    v = (v - mean[ch]) * rsqrtf(var[ch] + EPSV) * gam[ch] + bet[ch];
    if (res) v += res[i];
    out[i] = v;
}

// ---------------- pooling --------------------------------------------------
__global__ void zero_kernel(float* __restrict__ p, int n) {
    int i = blockIdx.x * blockDim.x + threadIdx.x;
    if (i < n) p[i] = 0.0f;
}

__global__ void __launch_bounds__(256) pool_kernel(const float* __restrict__ x,
                                                   const int* __restrict__ batch,
                                                   float* __restrict__ ssum,
                                                   float* __restrict__ cnt, int N) {
    int node = blockIdx.x * 8 + (threadIdx.x >> 5);
    int lane = threadIdx.x & 31;
    if (node >= N) return;
    int g = batch[node];
    const float2* xv = (const float2*)(x + (size_t)node * HDIM);
    float2 v = xv[lane];
    atomicAdd(&ssum[g * HDIM + lane * 2],     v.x);
    atomicAdd(&ssum[g * HDIM + lane * 2 + 1], v.y);
    if (lane == 0) atomicAdd(&cnt[g], 1.0f);
}

// ---------------- tiny MLP head (64 graphs, one thread each) ---------------
__global__ void head_kernel(const float* __restrict__ ssum, const float* __restrict__ cnt,
                            const float* __restrict__ hW1, const float* __restrict__ hb1,
                            const float* __restrict__ hg, const float* __restrict__ hbta,
                            const float* __restrict__ hmean, const float* __restrict__ hvar,
                            const float* __restrict__ hW2, const float* __restrict__ hb2,
                            float* __restrict__ out, int G) {
    int g = threadIdx.x;
    if (g >= G) return;
    float inv = 1.0f / fmaxf(cnt[g], 1.0f);
    float o = hb2[0];
    for (int j = 0; j < 32; ++j) {
        float s = hb1[j];
        for (int k = 0; k < 64; ++k) s += ssum[g * HDIM + k] * inv * hW1[k * 32 + j];
        s = fmaxf(s, 0.0f);
        s = (s - hmean[j]) * rsqrtf(hvar[j] + EPSV) * hg[j] + hbta[j];
        o += s * hW2[j];
    }
    out[g] = o;
}

// ---------------------------------------------------------------------------
extern "C" void kernel_launch(void* const* d_in, const int* in_sizes, int n_in,
                              void* d_out, int out_size, void* d_ws, size_t ws_size,
                              hipStream_t stream) {
    const float* x        = (const float*)d_in[0];
    const int*   ei       = (const int*)d_in[1];
    const int*   batch    = (const int*)d_in[2];
    const float* Wc       = (const float*)d_in[3];
    const float* bc       = (const float*)d_in[4];
    const float* bn_gamma = (const float*)d_in[5];
    const float* bn_beta  = (const float*)d_in[6];
    const float* bn_mean  = (const float*)d_in[7];
    const float* bn_var   = (const float*)d_in[8];
    const float* hW1      = (const float*)d_in[9];
    const float* hb1      = (const float*)d_in[10];
    const float* hbn_g    = (const float*)d_in[11];
    const float* hbn_b    = (const float*)d_in[12];
    const float* hbn_m    = (const float*)d_in[13];
    const float* hbn_v    = (const float*)d_in[14];
    const float* hW2      = (const float*)d_in[15];
    const float* hb2      = (const float*)d_in[16];

    const int N = in_sizes[0] / HDIM;     // 100000
    const int E = in_sizes[1] / 2;        // 1000000
    const int G = out_size;               // 64
    const size_t NF = (size_t)N * HDIM;   // node-feature elements

    // workspace layout (floats)
    float* ws   = (float*)d_ws;
    size_t off  = 0;
    float* dinv = ws + off; off += ((size_t)N + 128) & ~(size_t)127;
    float* bufA = ws + off; off += NF;
    float* bufB = ws + off; off += NF;
    float* bufC = ws + off; off += NF;    // scatter accumulator
    float* ssum = ws + off; off += (size_t)G * HDIM;
    float* cnt  = ws + off; off += G;
    (void)ws_size; (void)n_in;

    const int TPB = 256;
    const int nodeBlocks = (N + TPB - 1) / TPB;
    const int edgeBlocks = (E + TPB - 1) / TPB;
    const int elemBlocks = (int)((NF + TPB - 1) / TPB);
    const int gemmBlocks = (N + 127) / 128;
    const int edgeWaveBlocks = (E + 7) / 8;
    const int nodeWaveBlocks = (N + 7) / 8;

    // degree -> dinv
    deg_init_kernel<<<nodeBlocks, TPB, 0, stream>>>(dinv, N);
    deg_count_kernel<<<edgeBlocks, TPB, 0, stream>>>(ei, dinv, E);
    deg_rsqrt_kernel<<<nodeBlocks, TPB, 0, stream>>>(dinv, N);

    // 5 GCN layers with residuals, ping-ponging bufA/bufB
    float* bufs[2] = {bufA, bufB};
    const float* h = x;
    int xi = 0;  // buffer index holding current residual/x
    for (int l = 0; l < 5; ++l) {
        float* t          = (l == 0) ? bufs[1] : bufs[1 - xi];
        float* xnew       = (l == 0) ? bufs[0] : t;   // t is dead after scatter
        const float* res  = (l == 0) ? nullptr : bufs[xi];

        gemm_kernel<<<gemmBlocks, TPB, 0, stream>>>(h, Wc + (size_t)l * HDIM * HDIM, t, N);
        init_acc_kernel<<<elemBlocks, TPB, 0, stream>>>(t, dinv, bc + l * HDIM, bufC, (int)NF);
        scatter_kernel<<<edgeWaveBlocks, TPB, 0, stream>>>(t, ei, dinv, bufC, E);
        post_kernel<<<elemBlocks, TPB, 0, stream>>>(bufC, res,
                                                    bn_gamma + l * HDIM, bn_beta + l * HDIM,
                                                    bn_mean + l * HDIM, bn_var + l * HDIM,
                                                    xnew, (int)NF);
        h = xnew;
        if (l > 0) xi = 1 - xi;
    }
    // h == bufA here (x5)

    // global mean pool + head
    zero_kernel<<<(G * HDIM + G + TPB - 1) / TPB, TPB, 0, stream>>>(ssum, G * HDIM + G);
    pool_kernel<<<nodeWaveBlocks, TPB, 0, stream>>>(h, batch, ssum, cnt, N);
    head_kernel<<<1, 64, 0, stream>>>(ssum, cnt, hW1, hb1, hbn_g, hbn_b, hbn_m, hbn_v,
                                      hW2, hb2, (float*)d_out, G);
}